// RemoveBackground_8873402434071
// MI455X (gfx1250) — compile-verified
//
#include <hip/hip_runtime.h>
#include <stdint.h>

#if defined(__has_builtin)
#if __has_builtin(__builtin_amdgcn_tensor_load_to_lds)
#define HAVE_TDM 1
#endif
#endif

namespace {
constexpr int kW = 256;
constexpr int kH = 256;
constexpr int kT = 128;
constexpr int kC = 3;
constexpr int kTH = 32;             // interior rows per band
constexpr int kHALO = 5;
constexpr int kRB = kTH + 2*kHALO;  // 42 band rows (interior + halo)
constexpr int kNW = kW / 32;        // 8 mask words per 256-bit row
constexpr int kThreads = 256;
constexpr float kThresh = 0.5f;

constexpr int kXElems   = kC * kRB * kW;   // worst-case staged floats (126 KB)
constexpr int kLvlElems = 6 * kRB * kNW;   // dilation levels 0..5
constexpr int kFinElems = kTH * kNW;
constexpr size_t kSmemBytes =
    (size_t)kXElems*4 + (size_t)kLvlElems*4 + (size_t)kFinElems*4; // 138112 B

// Generic pointer to a __shared__ object: low 32 bits == byte offset within
// the workgroup's LDS allocation.
__device__ __forceinline__ uint32_t lds_off(const void* p) {
  return (uint32_t)(uintptr_t)p;
}

__device__ __forceinline__ void async_load_b128(uint32_t ldsoff, const void* gaddr) {
  asm volatile("global_load_async_to_lds_b128 %0, %1, off"
               :: "v"(ldsoff), "v"(gaddr)
               : "memory");
}
__device__ __forceinline__ void wait_async0() {
  asm volatile("s_wait_asynccnt 0" ::: "memory");
}

#if defined(HAVE_TDM)
typedef uint32_t u32x4 __attribute__((ext_vector_type(4)));
typedef int      i32x8 __attribute__((ext_vector_type(8)));
typedef int      i32x4 __attribute__((ext_vector_type(4)));

// One TDM descriptor moves the whole band: 3D tile, X=256 f32 cols, Y=nrows,
// Z=3 channels (Z stride = T*H*W elements). TDM packs LDS contiguously:
// exactly our [c][nrows][256] layout.
__device__ __forceinline__ void tdm_load_band(const float* gbase, uint32_t ldsoff,
                                              int nrows, uint64_t planeStrideElems) {
  uint64_t ga = (uint64_t)(uintptr_t)gbase;

  u32x4 g0;
  g0[0] = 1u;                                            // count=1 (valid), no gather
  g0[1] = ldsoff;                                        // lds_addr
  g0[2] = (uint32_t)ga;                                  // global_addr[31:0]
  g0[3] = ((uint32_t)(ga >> 32) & 0x01FFFFFFu)           // global_addr[56:32]
        | 0x80000000u;                                   // type=2 ("image")

  i32x8 g1;
  g1[0] = (int)(2u << 16);                               // wg_mask=0, data_size=4B
  g1[1] = (int)(256u << 16);                             // tensor_dim0[15:0]=256
  g1[2] = (int)(((uint32_t)nrows & 0xFFFFu) << 16);      // dim0 hi=0, tensor_dim1 lo
  g1[3] = (int)(256u << 16);                             // dim1 hi=0, tile_dim0=256
  g1[4] = (int)((uint32_t)nrows | (3u << 16));           // tile_dim1=nrows, tile_dim2=3
  g1[5] = 256;                                           // tensor_dim0_stride lo32
  g1[6] = 0;                                             // dim0_stride hi, dim1_stride lo16=0
  g1[7] = (int)(uint32_t)(planeStrideElems >> 16);       // tensor_dim1_stride[47:16]

  i32x4 g2;
  g2[0] = 3;                                             // tensor_dim2 = 3 channels
  g2[1] = 0; g2[2] = 0; g2[3] = 0;                       // tensor_dim3/dim2_stride/tile_dim3 = 0
  i32x4 g3;
  g3[0] = 0; g3[1] = 0; g3[2] = 0; g3[3] = 0;

  i32x8 g4;                                              // 5th operand on this
  g4[0] = 0; g4[1] = 0; g4[2] = 0; g4[3] = 0;            // toolchain (probe-
  g4[4] = 0; g4[5] = 0; g4[6] = 0; g4[7] = 0;            // verified zero-filled)

  __builtin_amdgcn_tensor_load_to_lds(g0, g1, g2, g3, g4, 0);
}
#endif
} // namespace

extern "C" __global__ __launch_bounds__(kThreads)
void rmbg_fused(const float* __restrict__ X,
                const float* __restrict__ bg,
                float* __restrict__ out)
{
  extern __shared__ float smem[];
  float*    xplane = smem;                                    // [3][nrows][256] f32
  uint32_t* lvl    = (uint32_t*)(smem + kXElems);             // [6][42][8]
  uint32_t* fin    = (uint32_t*)(smem + kXElems) + kLvlElems; // [32][8]

  const int band = blockIdx.x;
  const int t    = blockIdx.y;
  const int h0   = band * kTH;
  const int tid  = threadIdx.x;
  const int lane = tid & 31;
  const int wv   = tid >> 5;

  const size_t planeStride = (size_t)kT * kH * kW;  // X/out are [C,T,H,W]
  const size_t tOff        = (size_t)t * kH * kW;

  // Staged row window, clipped to the image (TDM never reads out of bounds).
  const int rowsStart = (h0 - kHALO) < 0 ? 0 : (h0 - kHALO);
  const int rowsEnd   = (h0 + kTH + kHALO) > kH ? kH : (h0 + kTH + kHALO);
  const int nrows     = rowsEnd - rowsStart;        // 37 or 42
  const int topClip   = rowsStart - (h0 - kHALO);   // 0 or 5
  const int cstride   = nrows * kW;                 // LDS floats per channel

  // ---- Phase A0: stage X band (3 channels x nrows x 256) into LDS.
#if defined(HAVE_TDM)
  if (wv == 0) {
    tdm_load_band(X + tOff + (size_t)rowsStart * kW, lds_off(xplane),
                  nrows, (uint64_t)planeStride);
#if __has_builtin(__builtin_amdgcn_s_wait_tensorcnt)
    __builtin_amdgcn_s_wait_tensorcnt(0);
#else
    asm volatile("s_wait_tensorcnt 0x0" ::: "memory");
#endif
  }
#else
  for (int rq = tid; rq < nrows * (kW/4); rq += kThreads) {
    int r  = rq >> 6;            // row (shift, no division)
    int q4 = rq & 63;            // float4 within row
    const float* gp = X + tOff + (size_t)(rowsStart + r)*kW + q4*4;
    #pragma unroll
    for (int c = 0; c < kC; ++c)
      async_load_b128(lds_off(&xplane[c*cstride + r*kW + q4*4]),
                      gp + (size_t)c*planeStride);
  }
  wait_async0();
#endif
  __syncthreads();

  // ---- Phase A1: threshold -> packed mask bits (level 0) via wave32 ballot.
  for (int r = wv; r < kRB; r += kThreads/32) {
    int h  = h0 - kHALO + r;
    bool rowok = (h >= 0) && (h < kH);
    int lr  = r - topClip;
    int lrc = lr < 0 ? 0 : (lr >= nrows ? nrows-1 : lr);   // safe LDS row
    int hc  = h < 0 ? 0 : (h >= kH ? kH-1 : h);            // safe bg row
    for (int g = 0; g < kNW; ++g) {
      int col = g*32 + lane;
      const float* bp = bg + ((size_t)hc*kW + col) * 3;    // bg is [H,W,C]
      float s = fabsf(xplane[0*cstride + lrc*kW + col] - bp[0])
              + fabsf(xplane[1*cstride + lrc*kW + col] - bp[1])
              + fabsf(xplane[2*cstride + lrc*kW + col] - bp[2]);
      uint32_t wm = __builtin_amdgcn_ballot_w32(rowok && (s > kThresh));
      if (lane == 0) lvl[(0*kRB + r)*kNW + g] = wm;
    }
  }
  __syncthreads();

  // ---- Phase B: horizontal dilation levels 1..5 (one 256-bit row / thread).
  if (tid < kRB) {
    uint32_t m[kNW];
    #pragma unroll
    for (int i = 0; i < kNW; ++i) m[i] = lvl[(0*kRB + tid)*kNW + i];
    #pragma unroll
    for (int k = 1; k <= 5; ++k) {
      uint32_t n[kNW];
      #pragma unroll
      for (int i = 0; i < kNW; ++i) {
        uint32_t v = m[i] | (m[i] << 1) | (m[i] >> 1);
        if (i > 0)     v |= m[i-1] >> 31;
        if (i < kNW-1) v |= m[i+1] << 31;
        n[i] = v;
      }
      #pragma unroll
      for (int i = 0; i < kNW; ++i) { m[i] = n[i]; lvl[(k*kRB + tid)*kNW + i] = n[i]; }
    }
  }
  __syncthreads();

  // ---- Phase B2: diamond combine: out(r) = OR_{d=-5..5} hdilate_{5-|d|}(r+d).
  {
    int rp = tid >> 3;        // interior row 0..31
    int g  = tid & 7;
    int ri = rp + kHALO;      // band row index
    uint32_t v = lvl[(5*kRB + ri)*kNW + g];
    #pragma unroll
    for (int d = 1; d <= 5; ++d) {
      int k = 5 - d;
      v |= lvl[(k*kRB + (ri - d))*kNW + g];
      v |= lvl[(k*kRB + (ri + d))*kNW + g];
    }
    fin[rp*kNW + g] = v;
  }
  __syncthreads();

  // ---- Phase C: masked multiply, b128 LDS reads + b128 coalesced stores.
  {
    int q4  = tid & 63;       // col/4
    int r0  = tid >> 6;       // 0..3
    int col = q4 * 4;
    int wd  = col >> 5;
    int sh  = col & 31;
    for (int rr = r0; rr < kTH; rr += 4) {
      uint32_t wm = fin[rr*kNW + wd] >> sh;   // bits 0..3 -> cols col..col+3
      int h  = h0 + rr;
      int lr = rr + kHALO - topClip;          // LDS row (always in range)
      #pragma unroll
      for (int c = 0; c < kC; ++c) {
        const float4 x = *(const float4*)&xplane[c*cstride + lr*kW + col];
        float4 o;
        o.x = (wm & 1u) ? x.x : 0.0f;
        o.y = (wm & 2u) ? x.y : 0.0f;
        o.z = (wm & 4u) ? x.z : 0.0f;
        o.w = (wm & 8u) ? x.w : 0.0f;
        *(float4*)(out + (size_t)c*planeStride + tOff + (size_t)h*kW + col) = o;
      }
    }
  }
}

extern "C" void kernel_launch(void* const* d_in, const int* in_sizes, int n_in,
                              void* d_out, int out_size, void* d_ws, size_t ws_size,
                              hipStream_t stream) {
  (void)in_sizes; (void)n_in; (void)d_ws; (void)ws_size; (void)out_size;
  const float* X  = (const float*)d_in[0];   // [3,128,256,256] f32
  const float* bg = (const float*)d_in[1];   // [256,256,3] f32
  float* out = (float*)d_out;                // [3,128,256,256] f32

  static_assert(kSmemBytes < 320u*1024u, "exceeds CDNA5 WGP LDS");
  (void)hipFuncSetAttribute((const void*)rmbg_fused,
                            hipFuncAttributeMaxDynamicSharedMemorySize,
                            (int)kSmemBytes);

  dim3 grid(kH / kTH, kT);   // 8 bands x 128 frames = 1024 workgroups
  rmbg_fused<<<grid, kThreads, kSmemBytes, stream>>>(X, bg, out);
}